// GraphConv_31318901522779
// MI455X (gfx1250) — compile-verified
//
#include <hip/hip_runtime.h>
#include <stdint.h>

#define N_NODES 50000
#define N_EDGES 1600000
#define D       512      // D_IN == D_OUT

typedef __attribute__((ext_vector_type(2))) float v2f;
typedef __attribute__((ext_vector_type(4))) float v4f;
typedef __attribute__((ext_vector_type(8))) float v8f;

// ---------------------------------------------------------------------------
// Kernel 1: out[n][d] = bias[d]   (bias folded in before the edge scatter)
// ---------------------------------------------------------------------------
__global__ void __launch_bounds__(256)
bias_init_kernel(const float* __restrict__ bias, float* __restrict__ out) {
    int idx = blockIdx.x * 256 + threadIdx.x;   // over N_NODES*128 float4 slots
    int d4  = idx & 127;                        // float4 index within a row
    v4f b   = ((const v4f*)bias)[d4];
    ((v4f*)out)[idx] = b;
}

// ---------------------------------------------------------------------------
// Kernel 2: hidden = X @ W  using V_WMMA_F32_16X16X4_F32 (full fp32 numerics)
//
// Block = 256 threads = 8 waves, one 16-row M-tile per block (50000 = 3125*16).
// The 16x512 X tile (32 KB, contiguous rows) is staged into LDS once per
// block via the CDNA5 async global->LDS path (ASYNCcnt), then all 8 waves
// read their A fragments from LDS (ds_load_b64) — X is read exactly once,
// coalesced, instead of 8x duplicated per-lane strided gathers.
//
// Wave w owns N range [64*w, 64*w+64): four 16x16 f32 accumulators, K in
// steps of 4.
//
// ISA fragment layouts (CDNA5 ISA):
//   A 16x4 f32 : lane = M (mod 16); per-lane v2f of K+koff, K+koff+1,
//                koff = 2*(lane/16).
//   B 4x16 f32 : VGPR0 lanes0-15 = row K+0, lanes16-31 = row K+2;
//                VGPR1 = rows K+1 / K+3; N = lane%16.
//   C/D 16x16  : VGPR j, lanes0-15 -> M=j, lanes16-31 -> M=j+8; N = lane%16.
// ---------------------------------------------------------------------------
__global__ void __launch_bounds__(256)
gemm_wmma_kernel(const float* __restrict__ X, const float* __restrict__ W,
                 float* __restrict__ H) {
    __shared__ float xs[16 * D];                // 32 KB M-tile stage

    const int tid  = threadIdx.x;
    const int lane = tid & 31;
    const int wave = tid >> 5;                  // 0..7 -> which 64-wide N strip
    const int m0   = blockIdx.x * 16;
    const int mrow = lane & 15;                 // A-matrix M for this lane
    const int koff = (lane >> 4) * 2;           // 0 or 2 (half-wave K offset)
    const int n0   = wave * 64;
    const int ncol = lane & 15;                 // B/C/D N for this lane

    // ---- async stage: 32 KB contiguous (16 rows x 512 f32) -> LDS ----------
    {
        const char* gbase = (const char*)(X + (size_t)m0 * D);
#pragma unroll
        for (int r = 0; r < 8; ++r) {
            uint32_t byte_off = (uint32_t)(tid * 16 + r * 4096);
            uint64_t gaddr    = (uint64_t)(uintptr_t)(gbase + byte_off);
            // LDS aperture rule: the low 32 bits of the flat address are the LDS offset
            uint32_t laddr    = (uint32_t)(uintptr_t)((char*)xs + byte_off);
            asm volatile("global_load_async_to_lds_b128 %0, %1, off"
                         :: "v"(laddr), "v"(gaddr) : "memory");
        }
        asm volatile("s_wait_asynccnt 0x0" ::: "memory");
    }
    __syncthreads();

    const float* __restrict__ xtile = xs + (size_t)mrow * D;

    v8f acc0 = {}, acc1 = {}, acc2 = {}, acc3 = {};

    for (int k = 0; k < D; k += 4) {
        // A fragment from LDS: two consecutive K values for this lane's row
        v2f a = *(const v2f*)(xtile + k + koff);

        // B fragments for the four 16-wide N tiles (W is L0/L2 resident)
        const float* wb = W + (size_t)(k + koff) * D + n0 + ncol;
        v2f b0, b1, b2, b3;
        b0.x = wb[0];      b0.y = wb[D];        // rows K+koff, K+koff+1
        b1.x = wb[16];     b1.y = wb[D + 16];
        b2.x = wb[32];     b2.y = wb[D + 32];
        b3.x = wb[48];     b3.y = wb[D + 48];

        acc0 = __builtin_amdgcn_wmma_f32_16x16x4_f32(false, a, false, b0,
                                                     (short)0, acc0, false, false);
        acc1 = __builtin_amdgcn_wmma_f32_16x16x4_f32(false, a, false, b1,
                                                     (short)0, acc1, false, false);
        acc2 = __builtin_amdgcn_wmma_f32_16x16x4_f32(false, a, false, b2,
                                                     (short)0, acc2, false, false);
        acc3 = __builtin_amdgcn_wmma_f32_16x16x4_f32(false, a, false, b3,
                                                     (short)0, acc3, false, false);
    }

    // Store D-matrix per C/D layout: VGPR j -> row m0 + j + 8*(lane/16)
    const int srow = m0 + ((lane >> 4) << 3);
    float* __restrict__ hbase = H + (size_t)srow * D + n0 + ncol;
#pragma unroll
    for (int j = 0; j < 8; ++j) {
        float* hj = hbase + (size_t)j * D;
        hj[0]  = acc0[j];
        hj[16] = acc1[j];
        hj[32] = acc2[j];
        hj[48] = acc3[j];
    }
}

// ---------------------------------------------------------------------------
// Kernel 3: edge scatter. out[row[e]] += val[e] * hidden[col[e]]
// 2 edges per 256-thread block; 128 lanes x float4 covers one 512-f32 row.
// hidden (102.4 MB) is L2-resident on MI455X (192 MB L2), so both the random
// gather and the fp32 atomics stay on-chip.
// ---------------------------------------------------------------------------
__global__ void __launch_bounds__(256)
scatter_kernel(const float* __restrict__ H, const float* __restrict__ edge_val,
               const int* __restrict__ edge_row, const int* __restrict__ edge_col,
               float* __restrict__ out) {
    const int e = blockIdx.x * 2 + (threadIdx.x >> 7);
    const int t = threadIdx.x & 127;

    const int   col = edge_col[e];
    const int   row = edge_row[e];
    const float val = edge_val[e];

    v4f h = ((const v4f*)(H + (size_t)col * D))[t];
    float* o = out + (size_t)row * D + (size_t)t * 4;
    atomicAdd(o + 0, val * h.x);
    atomicAdd(o + 1, val * h.y);
    atomicAdd(o + 2, val * h.z);
    atomicAdd(o + 3, val * h.w);
}

// ---------------------------------------------------------------------------
extern "C" void kernel_launch(void* const* d_in, const int* in_sizes, int n_in,
                              void* d_out, int out_size, void* d_ws, size_t ws_size,
                              hipStream_t stream) {
    (void)in_sizes; (void)n_in; (void)out_size; (void)ws_size;

    const float* x    = (const float*)d_in[0];  // [N_NODES, D]
    const float* w    = (const float*)d_in[1];  // [D, D]
    const float* bias = (const float*)d_in[2];  // [1, D]
    const float* eval = (const float*)d_in[3];  // [N_EDGES]
    const int*   erow = (const int*)d_in[4];    // [N_EDGES]
    const int*   ecol = (const int*)d_in[5];    // [N_EDGES]

    float* out    = (float*)d_out;              // [N_NODES, D]
    float* hidden = (float*)d_ws;               // [N_NODES, D] scratch (102.4 MB)

    // out = bias (broadcast)
    bias_init_kernel<<<(N_NODES * 128) / 256, 256, 0, stream>>>(bias, out);
    // hidden = X @ W   (fp32 WMMA, async-LDS staged A tiles)
    gemm_wmma_kernel<<<N_NODES / 16, 256, 0, stream>>>(x, w, hidden);
    // out += scatter(edge_val * hidden[col]) into rows
    scatter_kernel<<<N_EDGES / 2, 256, 0, stream>>>(hidden, eval, erow, ecol, out);
}